// Attention_42949672961503
// MI455X (gfx1250) — compile-verified
//
#include <hip/hip_runtime.h>
#include <hip/hip_bf16.h>
#include <math.h>

#define N_SENT 262144
#define N_BAGS 8192
#define DIM    690
#define N_CLS  53
#define KPAD   704     // 176 * 4, zero-padded K for the WMMA GEMM
#define NPAD   64      // 4 * 16,  zero-padded N for the WMMA GEMM

typedef __attribute__((ext_vector_type(2))) float v2f;
typedef __attribute__((ext_vector_type(8))) float v8f;

// ---------------------------------------------------------------------------
// Kernel 0: combined per-class vector cmb[c,d] = att[c,d] * rel[c,d]
// ---------------------------------------------------------------------------
__global__ void cmb_kernel(const float* __restrict__ att,
                           const float* __restrict__ rel,
                           float* __restrict__ cmb) {
    int i = blockIdx.x * blockDim.x + threadIdx.x;
    if (i < N_CLS * DIM) cmb[i] = att[i] * rel[i];
}

// ---------------------------------------------------------------------------
// Kernel 0b: pad rel_weight [53,690] -> relpad [64, 704] (zeros elsewhere)
// so the WMMA GEMM needs no bounds checks in its inner loop.
// ---------------------------------------------------------------------------
__global__ void pad_rel_kernel(const float* __restrict__ rel,
                               float* __restrict__ relpad) {
    int i = blockIdx.x * blockDim.x + threadIdx.x;
    if (i < NPAD * KPAD) {
        int n = i / KPAD;
        int k = i - n * KPAD;
        relpad[i] = (n < N_CLS && k < DIM) ? rel[n * DIM + k] : 0.f;
    }
}

// ---------------------------------------------------------------------------
// Kernel A: per-sentence attention logit.  One wave32 per sentence.
// logit[i] = dot(x[i,:], cmb[q_i,:])
// ---------------------------------------------------------------------------
__global__ void logit_kernel(const float* __restrict__ x,
                             const float* __restrict__ cmb,
                             const int*   __restrict__ query,
                             float*       __restrict__ logit) {
    int wave = threadIdx.x >> 5;          // 8 waves per 256-thread block
    int lane = threadIdx.x & 31;
    int i = blockIdx.x * 8 + wave;
    const float* xr = x   + (long)i * DIM;
    const float* cr = cmb + (long)query[i] * DIM;
    float p = 0.f;
    for (int d = lane; d < DIM; d += 32) p += xr[d] * cr[d];
    // wave32 butterfly reduction
    for (int off = 16; off > 0; off >>= 1) p += __shfl_xor(p, off, 32);
    if (lane == 0) logit[i] = p;
}

// ---------------------------------------------------------------------------
// Kernel B: per-bag segment softmax + weighted sum.  One 256-thread block per
// bag; sentences of a bag are contiguous [scope[b], scope[b+1]).
// Writes bag_pad with row stride KPAD, zero-filling columns 690..703.
// ---------------------------------------------------------------------------
__global__ void bag_kernel(const float* __restrict__ x,
                           const float* __restrict__ logit,
                           const int*   __restrict__ scope,
                           float*       __restrict__ bag_pad) {
    __shared__ float red[256];
    int b   = blockIdx.x;
    int s   = scope[b];
    int e   = scope[b + 1];
    int tid = threadIdx.x;

    // block-reduce max
    float m = -INFINITY;
    for (int i = s + tid; i < e; i += 256) m = fmaxf(m, logit[i]);
    red[tid] = m; __syncthreads();
    for (int off = 128; off > 0; off >>= 1) {
        if (tid < off) red[tid] = fmaxf(red[tid], red[tid + off]);
        __syncthreads();
    }
    m = red[0]; __syncthreads();

    // block-reduce sum of exp
    float sm = 0.f;
    for (int i = s + tid; i < e; i += 256) sm += __expf(logit[i] - m);
    red[tid] = sm; __syncthreads();
    for (int off = 128; off > 0; off >>= 1) {
        if (tid < off) red[tid] += red[tid + off];
        __syncthreads();
    }
    float denom = red[0];

    // weighted accumulation over the bag's rows (coalesced across threads)
    int d0 = tid, d1 = tid + 256, d2 = tid + 512;   // d0,d1 always < 690
    float a0 = 0.f, a1 = 0.f, a2 = 0.f;
    for (int i = s; i < e; ++i) {
        float w = __expf(logit[i] - m);
        const float* xr = x + (long)i * DIM;
        a0 += w * xr[d0];
        a1 += w * xr[d1];
        if (d2 < DIM) a2 += w * xr[d2];
    }
    float inv = 1.f / denom;
    float* br = bag_pad + (long)b * KPAD;
    br[d0] = a0 * inv;
    br[d1] = a1 * inv;
    if (d2 < KPAD) br[d2] = (d2 < DIM) ? a2 * inv : 0.f;  // zero K padding
}

// ---------------------------------------------------------------------------
// Kernel C: final classifier via WMMA F32 16x16x4 on padded operands.
// out[8192,53] = bag_pad[8192,704(=690+pad)] @ relpad[64,704]^T + bias
// One wave per 16x16 output tile: 512 M-tiles x 4 N-tiles.
// Inner loop: 176 iterations, two unconditional b64 loads + one WMMA.
// A layout (16x4): lanes 0-15 hold (K=0,1) of row M=lane%16, lanes 16-31 hold
// (K=2,3).  B layout mirrors with N in place of M.
// C/D layout: VGPR r, lane<16 -> (M=r, N=lane); lane>=16 -> (M=r+8, N=lane-16)
// ---------------------------------------------------------------------------
__global__ void gemm_kernel(const float* __restrict__ bag_pad,
                            const float* __restrict__ relpad,
                            const float* __restrict__ bias,
                            float*       __restrict__ out) {
    int tile = blockIdx.x;
    int m0   = (tile >> 2) * 16;
    int n0   = (tile & 3) * 16;
    int lane = threadIdx.x;        // 0..31, one wave per block
    int half = lane >> 4;          // 0: serves K=4t+{0,1}   1: K=4t+{2,3}
    int l    = lane & 15;
    int m    = m0 + l;             // A row served by this lane
    int n    = n0 + l;             // B column served by this lane

    // v2f element index for K-step t is 2*t + half  (float offset 4t + 2*half)
    const v2f* arow = (const v2f*)(bag_pad + (long)m * KPAD) + half;
    const v2f* brow = (const v2f*)(relpad  + (long)n * KPAD) + half;

    v8f acc = {};
#pragma unroll 4
    for (int t = 0; t < KPAD / 4; ++t) {
        v2f a = arow[2 * t];
        v2f bm = brow[2 * t];
        // (neg_a, A, neg_b, B, c_mod, C, reuse_a, reuse_b)
        acc = __builtin_amdgcn_wmma_f32_16x16x4_f32(
                  false, a, false, bm, (short)0, acc, false, false);
    }

    if (n < N_CLS) {
        float bv = bias[n];
#pragma unroll
        for (int r = 0; r < 8; ++r) {
            int row = m0 + r + half * 8;
            out[(long)row * N_CLS + n] = acc[r] + bv;
        }
    }
}

// ---------------------------------------------------------------------------
// Launch
// ---------------------------------------------------------------------------
extern "C" void kernel_launch(void* const* d_in, const int* in_sizes, int n_in,
                              void* d_out, int out_size, void* d_ws, size_t ws_size,
                              hipStream_t stream) {
    const float* x     = (const float*)d_in[0];   // [N_SENT, DIM]
    const float* rel   = (const float*)d_in[1];   // [N_CLS, DIM]
    const float* att   = (const float*)d_in[2];   // [N_CLS, DIM]
    const float* bias  = (const float*)d_in[3];   // [N_CLS]
    const int*   query = (const int*)d_in[4];     // [N_SENT]
    const int*   scope = (const int*)d_in[5];     // [N_BAGS+1]
    float* out = (float*)d_out;                   // [N_BAGS, N_CLS]

    // workspace layout (floats):
    //   [0, 36570)        cmb                  (padded region to 40960)
    //   [40960, +262144)  per-sentence logits
    //   [303104, +8192*704)  bag_pad  (zero-padded K)
    //   [6070272, +64*704)   relpad   (zero-padded rel_weight)
    //   total ~24.5 MB
    float* wsf      = (float*)d_ws;
    float* cmb      = wsf;
    float* logit    = wsf + 40960;
    float* bag_pad  = wsf + 40960 + N_SENT;
    float* relpad   = bag_pad + (long)N_BAGS * KPAD;

    cmb_kernel<<<(N_CLS * DIM + 255) / 256, 256, 0, stream>>>(att, rel, cmb);
    pad_rel_kernel<<<(NPAD * KPAD + 255) / 256, 256, 0, stream>>>(rel, relpad);
    logit_kernel<<<N_SENT / 8, 256, 0, stream>>>(x, cmb, query, logit);
    bag_kernel<<<N_BAGS, 256, 0, stream>>>(x, logit, scope, bag_pad);
    gemm_kernel<<<(N_BAGS / 16) * 4, 32, 0, stream>>>(bag_pad, relpad, bias, out);
}